// FeaAlign_39101382263356
// MI455X (gfx1250) — compile-verified
//
#include <hip/hip_runtime.h>
#include <math.h>

// ---------------------------------------------------------------------------
// FeaAlign pipeline for MI455X (gfx1250, wave32, WMMA).
// All GEMM-like work runs on V_WMMA_F32_16X16X4_F32 (fp32-exact vs reference).
// Inner GEMM loops are fed purely from LDS: weights are pre-permuted into the
// WMMA A-operand register order (one ds_load_b64 per operand), input halo
// tiles are staged with padding resolved, so the hot loop is ds + wmma only.
// Shapes fixed by the reference: B=4, C=32, H=W=192, DG=8, K=9.
// ---------------------------------------------------------------------------

typedef __attribute__((ext_vector_type(2))) float v2f;
typedef __attribute__((ext_vector_type(8))) float v8f;

#define BB   4
#define CC   32
#define HH   192
#define WW   192
#define HWSZ (HH * WW)            // 36864
#define NPIX (BB * HWSZ)          // 147456; W%16==0 -> a 16-tile is one row run
#define KDIM 288                  // C * 9 for 3x3 convs and the deform GEMM

__device__ __forceinline__ v8f wmma_f32(v2f a, v2f b, v8f c) {
  // 8 args: (neg_a, A, neg_b, B, c_mod, C, reuse_a, reuse_b)
  return __builtin_amdgcn_wmma_f32_16x16x4_f32(false, a, false, b, (short)0, c,
                                               false, false);
}

__device__ __forceinline__ float sigmoidf_(float v) {
  return 1.0f / (1.0f + __expf(-v));
}

// ---------------------------------------------------------------------------
// Implicit-GEMM conv (1x1 or 3x3, pad = KS/2). One wave computes a 32(M) x
// 16(N=pixels) tile, K stepped by 4 via f32 WMMA.
// WMMA f32 16x16x4 operand layouts (wave32):
//   A (16x4):  lanes 0-15 M=0..15 {K0,K1}; lanes 16-31 {K2,K3}
//   B (4x16):  lanes 0-15 N=0..15 {K0,K1}; lanes 16-31 {K2,K3}
//   D (16x16): vgpr r, lane<16 -> M=r, lane>=16 -> M=8+r; N = lane&15
// aLds holds weights already permuted into A-operand order:
//   aLds[(s*NITER + j)*64 + lane*2 + c] = wgt[mBase+s*16+(lane&15)]
//                                            [4j + (lane>>4)*2 + c]
// so the inner-loop A fetch is one conflict-free ds_load_b64 per subtile.
// ---------------------------------------------------------------------------
template <int CIN, int KS, int COUT, int WPB>
__global__ __launch_bounds__(WPB * 32) void conv_wmma_kernel(
    const float* __restrict__ in0, const float* __restrict__ in1, int split,
    const float* __restrict__ wgt,   // [COUT][CIN*KS*KS] row-major (OIHW flat)
    const float* __restrict__ bias, float* __restrict__ out, int sigFrom) {
  constexpr int KTOT  = CIN * KS * KS;
  constexpr int NITER = KTOT / 4;
  constexpr int INSZ  = (KS == 1) ? CIN * 16 : CIN * 54;  // 54 = 3 rows * 18

  __shared__ float aLds[2 * NITER * 64];  // permuted weights (whole block)
  __shared__ float sIn[WPB][INSZ];        // per-wave input tile (+halo)

  const int tid   = threadIdx.x;
  const int lane  = tid & 31;
  const int wid   = tid >> 5;
  const int nTile = blockIdx.x * WPB + wid;
  const int mBase = blockIdx.y * 32;

  // tile-uniform pixel decode (16-pixel run within one row)
  const int pbase = nTile * 16;
  const int b     = pbase / HWSZ;
  const int rem   = pbase - b * HWSZ;
  const int y     = rem / WW;
  const int xb    = rem - y * WW;

  __builtin_prefetch(wgt + (size_t)mBase * KTOT, 0, 0);
  __builtin_prefetch(in0 + ((size_t)(b * (KS == 1 ? split : CIN)) * HH + y) * WW + xb, 0, 0);

  // ---- stage weights permuted into WMMA A-operand order -------------------
  for (int e = tid; e < 2 * NITER * 64; e += WPB * 32) {
    const int s = e / (NITER * 64);
    const int r = e - s * (NITER * 64);
    const int j = r >> 6;
    const int q = r & 63;
    const int l = q >> 1;
    const int c = q & 1;
    const int row = mBase + s * 16 + (l & 15);
    const int k   = 4 * j + (l >> 4) * 2 + c;
    if constexpr (COUT % 32 == 0) {
      aLds[e] = wgt[(size_t)row * KTOT + k];
    } else {
      aLds[e] = (row < COUT) ? wgt[(size_t)row * KTOT + k] : 0.0f;
    }
  }

  // ---- stage input tile (zero-padding resolved here) ----------------------
  if constexpr (KS == 1) {
    for (int e = lane; e < CIN * 16; e += 32) {
      const int i = e >> 4;
      const int n = e & 15;
      float v;
      if (i < split) {
        v = in0[((size_t)(b * split + i) * HH + y) * WW + xb + n];
      } else {
        v = in1[((size_t)(b * (CIN - split) + (i - split)) * HH + y) * WW + xb + n];
      }
      sIn[wid][e] = v;
    }
  } else {
    for (int e = lane; e < CIN * 54; e += 32) {
      const int i  = e / 54;
      const int r2 = e - i * 54;
      const int ry = r2 / 18;
      const int cx = r2 - ry * 18;
      const int yy = y - 1 + ry;
      const int xx = xb - 1 + cx;
      const bool ok = (yy >= 0) && (yy < HH) && (xx >= 0) && (xx < WW);
      sIn[wid][e] = ok ? in0[((size_t)(b * CIN + i) * HH + yy) * WW + xx] : 0.0f;
    }
  }
  __syncthreads();

  // ---- main loop: ds_load + wmma only -------------------------------------
  const int kSel = lane >> 4;
  const int n    = lane & 15;
  v8f acc0 = {};
  v8f acc1 = {};
#pragma unroll 4
  for (int j = 0; j < NITER; ++j) {
    const v2f av0 = *(const v2f*)(aLds + j * 64 + lane * 2);
    const v2f av1 = *(const v2f*)(aLds + (NITER + j) * 64 + lane * 2);
    const int k0  = 4 * j + kSel * 2;
    v2f bv;
    if constexpr (KS == 1) {
      bv.x = sIn[wid][k0 * 16 + n];
      bv.y = sIn[wid][(k0 + 1) * 16 + n];
    } else {
      const int k1 = k0 + 1;
      const int i0 = k0 / 9, t0 = k0 - i0 * 9;
      const int i1 = k1 / 9, t1 = k1 - i1 * 9;
      bv.x = sIn[wid][i0 * 54 + (t0 / 3) * 18 + n + (t0 - (t0 / 3) * 3)];
      bv.y = sIn[wid][i1 * 54 + (t1 / 3) * 18 + n + (t1 - (t1 / 3) * 3)];
    }
    acc0 = wmma_f32(av0, bv, acc0);
    acc1 = wmma_f32(av1, bv, acc1);
  }

  // ---- store 16 rows x 1 column per lane ----------------------------------
  const int mOff = kSel * 8;
#pragma unroll
  for (int r = 0; r < 8; ++r) {
    const int o  = mBase + mOff + r;
    const int o2 = o + 16;
    if (COUT % 32 == 0 || o < COUT) {
      float v = acc0[r] + bias[o];
      if (o >= sigFrom) v = sigmoidf_(v);
      out[((size_t)(b * COUT + o) * HH + y) * WW + xb + n] = v;
    }
    if (COUT % 32 == 0 || o2 < COUT) {
      float v = acc1[r] + bias[o2];
      if (o2 >= sigFrom) v = sigmoidf_(v);
      out[((size_t)(b * COUT + o2) * HH + y) * WW + xb + n] = v;
    }
  }
}

// ---------------------------------------------------------------------------
// Modulated deformable conv v2. Per wave: build sampled[288][16] in LDS via
// bilinear gather (VALU), then GEMM (32 x 288) x (288 x 16) on f32 WMMA with
// 8-byte global A loads (k0,k0+1 contiguous, even offset -> global_load_b64).
// co layout: channels [0,72)=dy, [72,144)=dx, [144,216)=mask (pre-sigmoided).
// ---------------------------------------------------------------------------
__device__ __forceinline__ float bsample(const float* __restrict__ ch, int y,
                                         int x) {
  const bool ok = (y >= 0) && (y < HH) && (x >= 0) && (x < WW);
  const int  yi = min(max(y, 0), HH - 1);
  const int  xi = min(max(x, 0), WW - 1);
  const float v = ch[yi * WW + xi];
  return ok ? v : 0.0f;
}

template <int WPB>
__global__ __launch_bounds__(WPB * 32) void deform_wmma_kernel(
    const float* __restrict__ img,   // [B,32,H,W]
    const float* __restrict__ co,    // [B,216,H,W]
    const float* __restrict__ wgt,   // dcn_w as [32][288]
    const float* __restrict__ bias,  // [32]
    float* __restrict__ out) {       // [B,32,H,W]
  __shared__ float smp[WPB][KDIM * 16];  // 18KB per wave, conflict-free layout

  const int lane  = threadIdx.x & 31;
  const int wid   = threadIdx.x >> 5;
  const int nTile = blockIdx.x * WPB + wid;
  const int pbase = nTile * 16;

  // tile-uniform pixel decode
  const int b   = pbase / HWSZ;
  const int rem = pbase - b * HWSZ;
  const int y   = rem / WW;
  const int xb  = rem - y * WW;

  __builtin_prefetch(co + ((size_t)(b * 216) * HH + y) * WW + xb, 0, 0);

  // ---- phase 1: bilinear gather into LDS (288*16 / 32 = 144 iters) --------
  for (int it = 0; it < (KDIM * 16) / 32; ++it) {
    const int e  = it * 32 + lane;
    const int k  = e >> 4;       // 0..287 = i*9 + t
    const int n  = e & 15;
    const int i  = k / 9;        // input channel
    const int t  = k - i * 9;    // tap index
    const int dg = i >> 2;       // deform group (Cg = 4)
    const int x  = xb + n;

    const size_t cidx = ((size_t)(b * 216 + dg * 9 + t) * HH + y) * WW + x;
    const float dyv = co[cidx];
    const float dxv = co[cidx + (size_t)72 * HWSZ];
    const float mv  = co[cidx + (size_t)144 * HWSZ];  // sigmoid applied upstream

    const float pyf = dyv + (float)(t / 3 - 1) + (float)y;
    const float pxf = dxv + (float)(t - (t / 3) * 3 - 1) + (float)x;
    const float y0f = floorf(pyf);
    const float x0f = floorf(pxf);
    const float wy  = pyf - y0f;
    const float wx  = pxf - x0f;
    const int   y0  = (int)y0f;
    const int   x0  = (int)x0f;

    const float* ch = img + (size_t)(b * CC + i) * HWSZ;
    const float v00 = bsample(ch, y0, x0);
    const float v01 = bsample(ch, y0, x0 + 1);
    const float v10 = bsample(ch, y0 + 1, x0);
    const float v11 = bsample(ch, y0 + 1, x0 + 1);

    const float val =
        ((1.0f - wy) * ((1.0f - wx) * v00 + wx * v01) +
         wy * ((1.0f - wx) * v10 + wx * v11)) * mv;
    smp[wid][k * 16 + n] = val;
  }
  __syncthreads();

  // ---- phase 2: (32 x 288) x (288 x 16) via 72x2 f32 WMMAs ----------------
  const int m    = lane & 15;
  const int kSel = lane >> 4;
  const int ncol = lane & 15;

  v8f acc0 = {};
  v8f acc1 = {};
#pragma unroll 4
  for (int kk = 0; kk < KDIM; kk += 4) {
    const int k0 = kk + kSel * 2;
    v2f bv;
    bv.x = smp[wid][k0 * 16 + ncol];
    bv.y = smp[wid][(k0 + 1) * 16 + ncol];
    const v2f av0 = *(const v2f*)(wgt + (size_t)m * KDIM + k0);         // b64
    const v2f av1 = *(const v2f*)(wgt + (size_t)(16 + m) * KDIM + k0);  // b64
    acc0 = wmma_f32(av0, bv, acc0);
    acc1 = wmma_f32(av1, bv, acc1);
  }

  const int mOff = kSel * 8;
#pragma unroll
  for (int r = 0; r < 8; ++r) {
    const int o  = mOff + r;
    const int o2 = o + 16;
    out[((size_t)(b * CC + o) * HH + y) * WW + xb + ncol]  = acc0[r] + bias[o];
    out[((size_t)(b * CC + o2) * HH + y) * WW + xb + ncol] = acc1[r] + bias[o2];
  }
}

// ---------------------------------------------------------------------------
// Launcher. Workspace layout (floats):
//   [0, 32*NPIX)                   feat (live until deform's producer done)
//   [32*NPIX, 32*NPIX + 216*NPIX)  co (216 ch); img_cat and feat1 overlap this
//                                  region (both dead before co is written).
// Total ~146 MB of the scratch buffer.
// ---------------------------------------------------------------------------
extern "C" void kernel_launch(void* const* d_in, const int* in_sizes, int n_in,
                              void* d_out, int out_size, void* d_ws,
                              size_t ws_size, hipStream_t stream) {
  (void)in_sizes; (void)n_in; (void)out_size; (void)ws_size;
  const float* img_ref = (const float*)d_in[0];
  const float* img     = (const float*)d_in[1];
  const float* c1w  = (const float*)d_in[2];
  const float* c1b  = (const float*)d_in[3];
  const float* c2aw = (const float*)d_in[4];
  const float* c2ab = (const float*)d_in[5];
  const float* c2bw = (const float*)d_in[6];
  const float* c2bb = (const float*)d_in[7];
  const float* offw = (const float*)d_in[8];
  const float* offb = (const float*)d_in[9];
  const float* dcnw = (const float*)d_in[10];
  const float* dcnb = (const float*)d_in[11];
  float* out = (float*)d_out;
  float* ws  = (float*)d_ws;

  const size_t chN = (size_t)NPIX;
  float* feat    = ws;                       // 32*NPIX
  float* co      = ws + 32 * chN;            // 216*NPIX
  float* img_cat = co;                       // 32*NPIX (dead before co write)
  float* feat1   = co + 32 * chN;            // 32*NPIX (dead before co write)

  const int nTiles = NPIX / 16;  // 9216
  const int NOSIG  = 1 << 30;

  // conv1: 1x1, Cin=64 (split concat), Cout=32; 8 waves/block
  conv_wmma_kernel<64, 1, 32, 8><<<dim3(nTiles / 8, 1), dim3(256), 0, stream>>>(
      img_ref, img, 32, c1w, c1b, img_cat, NOSIG);
  // conv2a: 3x3, 32->32; 4 waves/block (64.5KB LDS budget)
  conv_wmma_kernel<32, 3, 32, 4><<<dim3(nTiles / 4, 1), dim3(128), 0, stream>>>(
      img_cat, nullptr, 32, c2aw, c2ab, feat1, NOSIG);
  // conv2b: 3x3, 32->32
  conv_wmma_kernel<32, 3, 32, 4><<<dim3(nTiles / 4, 1), dim3(128), 0, stream>>>(
      feat1, nullptr, 32, c2bw, c2bb, feat, NOSIG);
  // offset/mask conv: 3x3, 32->216, sigmoid fused on channels >=144 (mask)
  conv_wmma_kernel<32, 3, 216, 4><<<dim3(nTiles / 4, 7), dim3(128), 0, stream>>>(
      feat, nullptr, 32, offw, offb, co, 144);
  // deformable conv: gather -> LDS -> WMMA GEMM
  deform_wmma_kernel<2><<<dim3(nTiles / 2), dim3(64), 0, stream>>>(
      img, co, dcnw, dcnb, out);
}